// PointNetSetAbstraction_42219528520116
// MI455X (gfx1250) — compile-verified
//
#include <hip/hip_runtime.h>
#include <hip/hip_bf16.h>

#define B_ 8
#define N_ 16384
#define S_ 1024
#define K_ 32
#define D_ 64
#define CINP 68                         // 67 channels padded to 68 (multiple of 4)
#define PTOT (B_*S_*K_)                 // 262144 points through the MLP
#define NTILE (PTOT/16)                 // 16384 16-point tiles
#define NBLK_MLP (NTILE/4)              // 4 waves (tiles) per 128-thread block

typedef __attribute__((ext_vector_type(2))) float v2f;
typedef __attribute__((ext_vector_type(8))) float v8f;

// ---------------------------------------------------------------------------
// prep kernels
// ---------------------------------------------------------------------------
__global__ __launch_bounds__(256) void prep_xn(const float* __restrict__ xyz,
                                               float* __restrict__ xn) {
  int t = blockIdx.x * 256 + threadIdx.x;          // B*N threads
  int b = t >> 14, n = t & (N_ - 1);
  const float* X = xyz + (size_t)b * 3 * N_;
  float x = X[n], y = X[N_ + n], z = X[2 * N_ + n];
  xn[t] = x * x + y * y + z * z;
}

__global__ __launch_bounds__(256) void prep_gather(const float* __restrict__ xyz,
    const int* __restrict__ sidx, float* __restrict__ qpts, float* __restrict__ qn,
    float* __restrict__ out_newxyz, float* __restrict__ out_idx) {
  int t = blockIdx.x * 256 + threadIdx.x;          // B*S threads
  int b = t / S_, s = t % S_;
  int n0 = sidx[t];
  const float* X = xyz + (size_t)b * 3 * N_;
  float x = X[n0], y = X[N_ + n0], z = X[2 * N_ + n0];
  qpts[t * 3 + 0] = x; qpts[t * 3 + 1] = y; qpts[t * 3 + 2] = z;
  qn[t] = x * x + y * y + z * z;
  out_newxyz[((size_t)b * 3 + 0) * S_ + s] = x;
  out_newxyz[((size_t)b * 3 + 1) * S_ + s] = y;
  out_newxyz[((size_t)b * 3 + 2) * S_ + s] = z;
  out_idx[t] = (float)n0;                          // idx output as float value
}

__global__ __launch_bounds__(256) void prep_w0(const float* __restrict__ w0,
                                               float* __restrict__ w0p) {
  int t = blockIdx.x * 256 + threadIdx.x;          // 64*68 threads
  int oc = t / CINP, ic = t % CINP;
  w0p[t] = (ic < 67) ? w0[oc * 67 + ic] : 0.0f;
}

// ---------------------------------------------------------------------------
// kNN: one wave per query. Per-lane sorted top-32 in LDS (keys = orderable
// float bits << 32 | index -> ascending distance, lowest index on ties, which
// matches jax.lax.top_k ordering). Then 32-round wave argmin merge.
// ---------------------------------------------------------------------------
__global__ __launch_bounds__(128) void knn_kernel(const float* __restrict__ xyz,
    const float* __restrict__ qpts, const float* __restrict__ qn,
    const float* __restrict__ xn, int* __restrict__ idxb,
    float* __restrict__ gnorm, float* __restrict__ out_gxyz) {
  __shared__ unsigned long long keys[32 * 128];    // [slot][tid] -> no bank conflict
  __shared__ unsigned long long outk[4][32];
  const int tid = threadIdx.x, lane = tid & 31, wv = tid >> 5;
  const int q = blockIdx.x * 4 + wv;               // 0..B*S-1
  const int b = q / S_;
  const float qx = qpts[q * 3 + 0], qy = qpts[q * 3 + 1], qz = qpts[q * 3 + 2];
  const float qnn = qn[q];
  const float* X = xyz + (size_t)b * 3 * N_;
  const float* XN = xn + (size_t)b * N_;

  #pragma unroll
  for (int j = 0; j < 32; ++j) keys[j * 128 + tid] = ~0ull;

  for (int n = lane; n < N_; n += 32) {
    float dx = X[n], dy = X[N_ + n], dz = X[2 * N_ + n];
    float d = qnn + XN[n] - 2.0f * (qx * dx + qy * dy + qz * dz);
    unsigned u = __float_as_uint(d);
    u = (u & 0x80000000u) ? ~u : (u | 0x80000000u);  // order-preserving map
    unsigned long long key = ((unsigned long long)u << 32) | (unsigned)n;
    if (key < keys[31 * 128 + tid]) {              // insertion into sorted list
      int j = 31;
      while (j > 0 && keys[(j - 1) * 128 + tid] > key) {
        keys[j * 128 + tid] = keys[(j - 1) * 128 + tid];
        --j;
      }
      keys[j * 128 + tid] = key;
    }
  }

  int ptr = 0;                                     // merge: pop global min 32x
  for (int r = 0; r < 32; ++r) {
    unsigned long long v = (ptr < 32) ? keys[ptr * 128 + tid] : ~0ull;
    unsigned long long m = v;
    #pragma unroll
    for (int off = 1; off < 32; off <<= 1) {
      unsigned long long o = __shfl_xor(m, off, 32);
      m = (o < m) ? o : m;
    }
    if (v == m) { ++ptr; outk[wv][r] = m; }        // keys unique -> one winner
  }
  __syncthreads();

  unsigned long long kk = outk[wv][lane];
  int n = (int)(kk & 0xffffffffull);
  int p = q * K_ + lane;
  idxb[p] = n;
  float gx = X[n], gy = X[N_ + n], gz = X[2 * N_ + n];
  out_gxyz[(size_t)p * 3 + 0] = gx;
  out_gxyz[(size_t)p * 3 + 1] = gy;
  out_gxyz[(size_t)p * 3 + 2] = gz;
  gnorm[(size_t)p * 3 + 0] = gx - qx;
  gnorm[(size_t)p * 3 + 1] = gy - qy;
  gnorm[(size_t)p * 3 + 2] = gz - qz;
}

// ---------------------------------------------------------------------------
// WMMA tile: D(16 oc x 16 pts) += W(16x4) * X(4x16), K-loop over channels.
// A layout: lane = M row, VGPR+half = K.  B layout: lane = N col, VGPR+half = K.
// C layout: lane = N col, VGPR r -> M = (lane/16)*8 + r.
// ---------------------------------------------------------------------------
__device__ __forceinline__ v8f wmma_tile(const float* __restrict__ W,
                                         const float* X, int ICp, int moff, int lane) {
  const int hf = lane >> 4;
  const int rw = lane & 15;
  const float* wrow = W + (size_t)(moff + rw) * ICp + hf * 2;
  const float* xrow = X + rw * ICp + hf * 2;
  v8f acc = {};
  for (int kc = 0; kc < ICp; kc += 4) {
    v2f a; a.x = wrow[kc]; a.y = wrow[kc + 1];
    v2f b; b.x = xrow[kc]; b.y = xrow[kc + 1];
    acc = __builtin_amdgcn_wmma_f32_16x16x4_f32(false, a, false, b, (short)0,
                                                acc, false, false);
  }
  return acc;
}

// deterministic per-block channel stats (sum, sumsq) of y = acc + bias
template <int NMT>
__device__ __forceinline__ void emit_stats(const v8f* acc, const float* __restrict__ bias,
    float* lstat, int wv, int hf, int pt, float* __restrict__ psum,
    float* __restrict__ psq, int C, int tid, int bx) {
  #pragma unroll
  for (int mt = 0; mt < NMT; ++mt) {
    #pragma unroll
    for (int r = 0; r < 8; ++r) {
      int ch = mt * 16 + hf * 8 + r;
      float y = acc[mt][r] + bias[ch];
      float s1 = y, s2 = y * y;
      #pragma unroll
      for (int o = 1; o < 16; o <<= 1) {           // reduce over 16 point lanes
        s1 += __shfl_xor(s1, o, 32);
        s2 += __shfl_xor(s2, o, 32);
      }
      if (pt == 0) {
        lstat[(wv * 2 + 0) * 128 + ch] = s1;
        lstat[(wv * 2 + 1) * 128 + ch] = s2;
      }
    }
  }
  __syncthreads();
  if (tid < C) {
    float s1 = lstat[0 * 128 + tid] + lstat[2 * 128 + tid] +
               lstat[4 * 128 + tid] + lstat[6 * 128 + tid];
    float s2 = lstat[1 * 128 + tid] + lstat[3 * 128 + tid] +
               lstat[5 * 128 + tid] + lstat[7 * 128 + tid];
    psum[(size_t)bx * 128 + tid] = s1;
    psq [(size_t)bx * 128 + tid] = s2;
  }
}

template <int NMT>
__device__ __forceinline__ void bn_relu_store(const v8f* acc, const float* __restrict__ bias,
    const float* __restrict__ sc, const float* __restrict__ sh, float* Y, int hf, int pt) {
  #pragma unroll
  for (int mt = 0; mt < NMT; ++mt)
    #pragma unroll
    for (int r = 0; r < 8; ++r) {
      int ch = mt * 16 + hf * 8 + r;
      float y = acc[mt][r] + bias[ch];
      Y[pt * 64 + ch] = fmaxf(fmaf(y, sc[ch], sh[ch]), 0.0f);
    }
}

// ---------------------------------------------------------------------------
// MLP chain. DEPTH=0/1/2: produce per-block partial BN stats of layer-DEPTH
// pre-BN output. DEPTH=3: full chain + relu + max over K -> new_points.
// ---------------------------------------------------------------------------
template <int DEPTH>
__global__ __launch_bounds__(128) void mlp_kernel(
    const float* __restrict__ pts, const int* __restrict__ idx,
    const float* __restrict__ gnorm, const float* __restrict__ w0p,
    const float* __restrict__ b0, const float* __restrict__ sc0, const float* __restrict__ sh0,
    const float* __restrict__ w1, const float* __restrict__ b1,
    const float* __restrict__ sc1, const float* __restrict__ sh1,
    const float* __restrict__ w2, const float* __restrict__ b2,
    const float* __restrict__ sc2, const float* __restrict__ sh2,
    float* __restrict__ psum, float* __restrict__ psq, float* __restrict__ out_np) {
  __shared__ float XA[4][16 * CINP];
  __shared__ float XB[4][16 * 64];
  __shared__ float lstat[4 * 2 * 128];
  __shared__ unsigned gmax[2][128];

  const int tid = threadIdx.x, lane = tid & 31, wv = tid >> 5;
  const int hf = lane >> 4, pt = lane & 15;
  const int tile = blockIdx.x * 4 + wv;
  const int p0 = tile * 16;
  const int b = (p0 / K_) / S_;
  const int p = p0 + pt;
  const int n = idx[p];

  if (DEPTH == 3) { gmax[0][tid] = 0u; gmax[1][tid] = 0u; }

  // build padded feature tile [16 pts][68 ch]: 3 norm-xyz + 64 point features
  float* X = XA[wv];
  for (int c = hf * 34; c < hf * 34 + 34; ++c) {
    float v;
    if (c < 3)            v = gnorm[(size_t)p * 3 + c];
    else if (c < 3 + D_)  v = pts[((size_t)b * D_ + (c - 3)) * N_ + n];
    else                  v = 0.0f;
    X[pt * CINP + c] = v;
  }
  __syncthreads();

  // ---- layer 0: 67->64 ----
  v8f acc0[4];
  #pragma unroll
  for (int mt = 0; mt < 4; ++mt) acc0[mt] = wmma_tile(w0p, X, CINP, mt * 16, lane);

  if (DEPTH == 0) {
    emit_stats<4>(acc0, b0, lstat, wv, hf, pt, psum, psq, 64, tid, blockIdx.x);
    return;
  }

  float* Y = XB[wv];
  bn_relu_store<4>(acc0, b0, sc0, sh0, Y, hf, pt);
  __syncthreads();

  // ---- layer 1: 64->64 ----
  v8f acc1[4];
  #pragma unroll
  for (int mt = 0; mt < 4; ++mt) acc1[mt] = wmma_tile(w1, Y, 64, mt * 16, lane);

  if (DEPTH == 1) {
    emit_stats<4>(acc1, b1, lstat, wv, hf, pt, psum, psq, 64, tid, blockIdx.x);
    return;
  }

  __syncthreads();                                  // done reading Y
  bn_relu_store<4>(acc1, b1, sc1, sh1, Y, hf, pt);  // overwrite with x2
  __syncthreads();

  // ---- layer 2: 64->128 ----
  v8f acc2[8];
  #pragma unroll
  for (int mt = 0; mt < 8; ++mt) acc2[mt] = wmma_tile(w2, Y, 64, mt * 16, lane);

  if (DEPTH == 2) {
    emit_stats<8>(acc2, b2, lstat, wv, hf, pt, psum, psq, 128, tid, blockIdx.x);
    return;
  }

  // ---- DEPTH == 3: relu(bn2) then max over K (relu >= 0 -> uint max) ----
  const int wg = wv >> 1;                           // 2 tiles (32 k) per group
  #pragma unroll
  for (int mt = 0; mt < 8; ++mt)
    #pragma unroll
    for (int r = 0; r < 8; ++r) {
      int ch = mt * 16 + hf * 8 + r;
      float y = acc2[mt][r] + b2[ch];
      float x = fmaxf(fmaf(y, sc2[ch], sh2[ch]), 0.0f);
      atomicMax(&gmax[wg][ch], __float_as_uint(x));
    }
  __syncthreads();
  for (int e = tid; e < 256; e += 128) {
    int g = e >> 7, c = e & 127;
    int group = blockIdx.x * 2 + g;                 // = b*S + s
    int bb = group / S_, ss = group % S_;
    out_np[((size_t)bb * 128 + c) * S_ + ss] = __uint_as_float(gmax[g][c]);
  }
}

// fold BN (training-mode batch stats) into scale/shift
__global__ __launch_bounds__(128) void bn_fin(const float* __restrict__ psum,
    const float* __restrict__ psq, const float* __restrict__ g,
    const float* __restrict__ be, float* __restrict__ sc, float* __restrict__ sh, int C) {
  int c = threadIdx.x;
  if (c >= C) return;
  float s1 = 0.0f, s2 = 0.0f;
  for (int blk = 0; blk < NBLK_MLP; ++blk) {        // fixed-order reduction
    s1 += psum[(size_t)blk * 128 + c];
    s2 += psq [(size_t)blk * 128 + c];
  }
  const float inv = 1.0f / (float)PTOT;
  float mu = s1 * inv;
  float var = s2 * inv - mu * mu;
  float r = rsqrtf(var + 1e-5f);
  float scale = g[c] * r;
  sc[c] = scale;
  sh[c] = fmaf(-mu, scale, be[c]);
}

// ---------------------------------------------------------------------------
extern "C" void kernel_launch(void* const* d_in, const int* in_sizes, int n_in,
                              void* d_out, int out_size, void* d_ws, size_t ws_size,
                              hipStream_t stream) {
  (void)in_sizes; (void)n_in; (void)out_size; (void)ws_size;
  const float* xyz  = (const float*)d_in[0];
  const float* pts  = (const float*)d_in[1];
  const int*   sidx = (const int*)d_in[2];
  const float* w0 = (const float*)d_in[3];
  const float* b0 = (const float*)d_in[4];
  const float* g0 = (const float*)d_in[5];
  const float* be0 = (const float*)d_in[6];
  const float* w1 = (const float*)d_in[7];
  const float* b1 = (const float*)d_in[8];
  const float* g1 = (const float*)d_in[9];
  const float* be1 = (const float*)d_in[10];
  const float* w2 = (const float*)d_in[11];
  const float* b2 = (const float*)d_in[12];
  const float* g2 = (const float*)d_in[13];
  const float* be2 = (const float*)d_in[14];

  float* out = (float*)d_out;                       // tuple concatenated flat
  float* out_newxyz = out;                          // [B,3,S]     24576
  float* out_np     = out + 24576;                  // [B,128,S]   1048576
  float* out_gxyz   = out + 24576 + 1048576;        // [B,S,K,3]   786432
  float* out_idx    = out + 24576 + 1048576 + 786432;  // [B,S]    8192

  float* ws   = (float*)d_ws;
  float* qpts = ws + 0;        // B*S*3
  float* qn   = ws + 24576;    // B*S
  float* xn   = ws + 32768;    // B*N
  float* gnorm = ws + 163840;  // B*S*K*3
  float* w0p  = ws + 950272;   // 64*68 padded weights
  float* sc0  = ws + 954624;  float* sh0 = ws + 954688;
  float* sc1  = ws + 954752;  float* sh1 = ws + 954816;
  float* sc2  = ws + 954880;  float* sh2 = ws + 955008;
  float* psum = ws + 955136;   // NBLK_MLP*128
  float* psq  = ws + 1479424;  // NBLK_MLP*128
  int*   idxb = (int*)(ws + 2003712);  // B*S*K ints

  prep_xn    <<<512, 256, 0, stream>>>(xyz, xn);
  prep_gather<<<32, 256, 0, stream>>>(xyz, sidx, qpts, qn, out_newxyz, out_idx);
  prep_w0    <<<17, 256, 0, stream>>>(w0, w0p);
  knn_kernel <<<(B_ * S_) / 4, 128, 0, stream>>>(xyz, qpts, qn, xn, idxb, gnorm, out_gxyz);

  mlp_kernel<0><<<NBLK_MLP, 128, 0, stream>>>(pts, idxb, gnorm, w0p, b0, sc0, sh0,
      w1, b1, sc1, sh1, w2, b2, sc2, sh2, psum, psq, out_np);
  bn_fin<<<1, 128, 0, stream>>>(psum, psq, g0, be0, sc0, sh0, 64);
  mlp_kernel<1><<<NBLK_MLP, 128, 0, stream>>>(pts, idxb, gnorm, w0p, b0, sc0, sh0,
      w1, b1, sc1, sh1, w2, b2, sc2, sh2, psum, psq, out_np);
  bn_fin<<<1, 128, 0, stream>>>(psum, psq, g1, be1, sc1, sh1, 64);
  mlp_kernel<2><<<NBLK_MLP, 128, 0, stream>>>(pts, idxb, gnorm, w0p, b0, sc0, sh0,
      w1, b1, sc1, sh1, w2, b2, sc2, sh2, psum, psq, out_np);
  bn_fin<<<1, 128, 0, stream>>>(psum, psq, g2, be2, sc2, sh2, 128);
  mlp_kernel<3><<<NBLK_MLP, 128, 0, stream>>>(pts, idxb, gnorm, w0p, b0, sc0, sh0,
      w1, b1, sc1, sh1, w2, b2, sc2, sh2, psum, psq, out_np);
}